// PostProcessor_9045201125701
// MI455X (gfx1250) — compile-verified
//
#include <hip/hip_runtime.h>
#include <math.h>

#define BATCH 4
#define NPROP 2048
#define NCLS 81
#define ROWS (BATCH * NPROP)
#define REGSTRIDE (NCLS * 8)
#define IMG_W 1024.0f
#define IMG_H 800.0f
#define SCORE_T 0.05f
#define NMS_T 0.5f
#define DETS 100
#define BBOX_CLIP 4.135166556742356f /* log(1000/16) */

typedef float v2f __attribute__((ext_vector_type(2)));
typedef float v8f __attribute__((ext_vector_type(8)));

// ---------------------------------------------------------------------------
// Box decode (maskrcnn BoxCoder, weights 10,10,5,5, TO_REMOVE=1) + clip
// ---------------------------------------------------------------------------
__device__ inline float4 decode_clip(float dx, float dy, float dw, float dh,
                                     float4 p) {
  float w = p.z - p.x + 1.0f;
  float h = p.w - p.y + 1.0f;
  float cx = p.x + 0.5f * w;
  float cy = p.y + 0.5f * h;
  dx *= 0.1f;                         // /WX
  dy *= 0.1f;                         // /WY
  dw = fminf(dw * 0.2f, BBOX_CLIP);   // /WW, clipped
  dh = fminf(dh * 0.2f, BBOX_CLIP);   // /WH, clipped
  float pcx = dx * w + cx;
  float pcy = dy * h + cy;
  float pw = expf(dw) * w;
  float ph = expf(dh) * h;
  float4 r;
  r.x = fminf(fmaxf(pcx - 0.5f * pw, 0.0f), IMG_W - 1.0f);
  r.y = fminf(fmaxf(pcy - 0.5f * ph, 0.0f), IMG_H - 1.0f);
  r.z = fminf(fmaxf(pcx + 0.5f * pw - 1.0f, 0.0f), IMG_W - 1.0f);
  r.w = fminf(fmaxf(pcy + 0.5f * ph - 1.0f, 0.0f), IMG_H - 1.0f);
  return r;
}

// ---------------------------------------------------------------------------
// Kernel 1: one wave32 per row. Softmax max/sum + argmax via shuffle
// butterfly; score at argmax == 1/sum(exp(l - max)). Lane 0 decodes only the
// selected class's 8 regression values (avoids streaming 21MB of regression).
// ---------------------------------------------------------------------------
__global__ void decode_score_kernel(const float* __restrict__ logits,
                                    const float* __restrict__ reg,
                                    const float4* __restrict__ lprop,
                                    const float4* __restrict__ rprop,
                                    float4* __restrict__ lbox,
                                    float4* __restrict__ rbox,
                                    float* __restrict__ score,
                                    float* __restrict__ skey,
                                    int* __restrict__ valid) {
  const int lane = threadIdx.x & 31;
  const int wave = threadIdx.x >> 5;
  const int row = blockIdx.x * 8 + wave;
  if (row >= ROWS) return;

  const float* lrow = logits + (size_t)row * NCLS;
  float v0 = lrow[lane];            // c = lane        (always < 81)
  float v1 = lrow[lane + 32];       // c = lane+32     (always < 81)
  float v2 = (lane + 64 < NCLS) ? lrow[lane + 64] : -INFINITY;

  // per-lane argmax, ascending c with strict '>' keeps first occurrence
  float bestV = v0;
  int bestC = lane;
  if (v1 > bestV) { bestV = v1; bestC = lane + 32; }
  if (v2 > bestV) { bestV = v2; bestC = lane + 64; }

  // wave argmax: max value, min index on ties (== jnp.argmax semantics)
  for (int off = 16; off >= 1; off >>= 1) {
    float ov = __shfl_xor(bestV, off);
    int oc = __shfl_xor(bestC, off);
    if (ov > bestV || (ov == bestV && oc < bestC)) { bestV = ov; bestC = oc; }
  }
  const float m = bestV;

  float z = expf(v0 - m) + expf(v1 - m);
  if (lane + 64 < NCLS) z += expf(v2 - m);
  for (int off = 16; off >= 1; off >>= 1) z += __shfl_xor(z, off);

  const float sc = 1.0f / z;  // exp(l_argmax - m) == 1
  const int vld = (bestC >= 1) && (sc > SCORE_T);

  if (lane == 0) {
    const float* rg = reg + (size_t)row * REGSTRIDE + bestC * 8;
    float4 lb = decode_clip(rg[0], rg[1], rg[2], rg[3], lprop[row]);
    float4 rb = decode_clip(rg[4], rg[5], rg[6], rg[7], rprop[row]);
    lbox[row] = lb;
    rbox[row] = rb;
    score[row] = sc;
    skey[row] = vld ? sc : -INFINITY;
    valid[row] = vld;
  }
}

// ---------------------------------------------------------------------------
// Kernel 2: per-image bitonic argsort in LDS (2048 keys, 1024 threads).
// Order: descending score, ascending index on ties == stable argsort(-s).
// On exit, also gathers both box arrays into sorted order so the N^2 stage
// reads contiguous, pipelineable b128 streams instead of dependent gathers.
// ---------------------------------------------------------------------------
__global__ void sort_kernel(const float* __restrict__ skey,
                            const float4* __restrict__ lbox,
                            const float4* __restrict__ rbox,
                            int* __restrict__ order,
                            float4* __restrict__ sortedL,
                            float4* __restrict__ sortedR) {
  __shared__ float k[NPROP];
  __shared__ int id[NPROP];
  const int img = blockIdx.x;
  const int t = threadIdx.x;  // 0..1023
  const float* sk = skey + (size_t)img * NPROP;
  k[t] = sk[t];
  id[t] = t;
  k[t + 1024] = sk[t + 1024];
  id[t + 1024] = t + 1024;
  __syncthreads();

  for (int kk = 2; kk <= NPROP; kk <<= 1) {
    for (int j = kk >> 1; j > 0; j >>= 1) {
      for (int p = t; p < NPROP; p += 1024) {
        int l = p ^ j;
        if (l > p) {
          bool up = ((p & kk) == 0);
          float ka = k[p], kb = k[l];
          int ia = id[p], ib = id[l];
          // lessLP: elem[l] sorts earlier than elem[p]
          bool lessLP = (kb > ka) || (kb == ka && ib < ia);
          bool doSwap = up ? lessLP : !lessLP;  // strict total order (idx unique)
          if (doSwap) {
            k[p] = kb; k[l] = ka;
            id[p] = ib; id[l] = ia;
          }
        }
      }
      __syncthreads();
    }
  }

  const size_t base = (size_t)img * NPROP;
  for (int p = t; p < NPROP; p += 1024) {
    const int idx = id[p];
    order[base + p] = idx;
    sortedL[base + p] = lbox[base + idx];
    sortedR[base + p] = rbox[base + idx];
  }
}

// ---------------------------------------------------------------------------
// Kernel 3: suppression bit-matrix. One wave per 16-row strip per
// (image,side) matrix. Union base area_i+area_j computed per 16x16 tile on
// the matrix pipe: D = A(16x4) x B(4x16), A=[area_i|1|0|0], B=[1;area_j;0;0]
// via V_WMMA_F32_16X16X4_F32. WMMA C layout (lane L, vgpr v -> M=v+8*(L>=16),
// N=L&15) lines up with the per-lane intersection math and __ballot packing.
// Below-diagonal tiles (j <= i guaranteed) are zero-filled, not computed.
// Boxes are pre-gathered in sorted order -> single independent b128 load per
// tile; IoU threshold is multiply-compare (uni > 0 always), no division.
// ---------------------------------------------------------------------------
__global__ void supp_kernel(const float4* __restrict__ sortedL,
                            const float4* __restrict__ sortedR,
                            unsigned short* __restrict__ supp) {
  const int lane = threadIdx.x & 31;
  const int wave = threadIdx.x >> 5;
  const int mi = blockIdx.x;  // 0..7 = img*2 + side
  const int img = mi >> 1;
  const int side = mi & 1;
  const int strip = blockIdx.y * 8 + wave;  // 0..127
  const int iBase = strip * 16;

  const float4* boxes =
      (side ? sortedR : sortedL) + (size_t)img * NPROP;

  // lane (L&15) owns sorted row iBase + (L&15); halves hold duplicates
  const float4 rbx = boxes[iBase + (lane & 15)];
  const float rArea = fmaxf(rbx.z - rbx.x + 1.0f, 0.0f) *
                      fmaxf(rbx.w - rbx.y + 1.0f, 0.0f);

  // Hoist row-box broadcasts (invariant across column tiles): this lane's
  // 8 WMMA-C rows are m = v + 8*(lane>=16).
  float rX1[8], rY1[8], rX2[8], rY2[8];
#pragma unroll
  for (int v = 0; v < 8; ++v) {
    const int src = v + 8 * (lane >> 4);
    rX1[v] = __shfl(rbx.x, src);
    rY1[v] = __shfl(rbx.y, src);
    rX2[v] = __shfl(rbx.z, src);
    rY2[v] = __shfl(rbx.w, src);
  }

  // A layout: lanes 0-15 hold K=0,1 ; lanes 16-31 hold K=2,3 (zeros)
  v2f a;
  if (lane < 16) { a.x = rArea; a.y = 1.0f; }
  else           { a.x = 0.0f;  a.y = 0.0f; }

  unsigned short* srow = supp + (size_t)mi * NPROP * 128;

  // Below-diagonal tiles are identically zero (predicate requires j > i):
  // zero-fill instead of computing. Lane L<16 owns row iBase+L.
  if (lane < 16) {
    unsigned short* rowp = srow + (size_t)(iBase + lane) * 128;
    for (int jt = 0; jt < strip; ++jt) rowp[jt] = 0;
  }

  for (int jt = strip; jt < 128; ++jt) {
    const float4 cbx = boxes[jt * 16 + (lane & 15)];
    const float cArea = fmaxf(cbx.z - cbx.x + 1.0f, 0.0f) *
                        fmaxf(cbx.w - cbx.y + 1.0f, 0.0f);

    // B layout: VGPR0 = rows K=0 (lanes 0-15) / K=2 (lanes 16-31), VGPR1 = K=1/K=3
    v2f b;
    if (lane < 16) { b.x = 1.0f; b.y = cArea; }
    else           { b.x = 0.0f; b.y = 0.0f; }

    v8f c = {};
    // D[m][n] = area_i[m] + area_j[n]
    c = __builtin_amdgcn_wmma_f32_16x16x4_f32(false, a, false, b, (short)0, c,
                                              false, false);

    const int j = jt * 16 + (lane & 15);
#pragma unroll
    for (int v = 0; v < 8; ++v) {
      const float ww = fminf(rX2[v], cbx.z) - fmaxf(rX1[v], cbx.x) + 1.0f;
      const float hh = fminf(rY2[v], cbx.w) - fmaxf(rY1[v], cbx.y) + 1.0f;
      const float inter = fmaxf(ww, 0.0f) * fmaxf(hh, 0.0f);
      const float uni = fmaxf(c[v] - inter, 1e-6f);
      const int i = iBase + v + 8 * (lane >> 4);
      // uni > 0 always, so inter/uni > T  <=>  inter > T*uni
      const bool pr = (inter > NMS_T * uni) && (j > i);
      const unsigned long long msk = __ballot(pr);
      if (lane == 0) {
        srow[(size_t)(iBase + v) * 128 + jt] =
            (unsigned short)(msk & 0xFFFFull);
        srow[(size_t)(iBase + v + 8) * 128 + jt] =
            (unsigned short)((msk >> 16) & 0xFFFFull);
      }
    }
  }
}

// ---------------------------------------------------------------------------
// Kernel 4: greedy NMS serial scan over precomputed bit-matrix. One 2-wave
// block per matrix; removed mask (64 words) lives in LDS; per step: 1 bit
// test + 64-word OR.
// ---------------------------------------------------------------------------
__global__ void nms_scan_kernel(const unsigned short* __restrict__ supp,
                                unsigned int* __restrict__ keepSorted) {
  __shared__ unsigned int removed[64];
  const int t = threadIdx.x;  // 0..63
  const int mi = blockIdx.x;
  removed[t] = 0u;
  __syncthreads();
  const unsigned short* sm = supp + (size_t)mi * NPROP * 128;
  for (int i = 0; i < NPROP; ++i) {
    const unsigned int rw = removed[i >> 5];
    __syncthreads();
    if (!((rw >> (i & 31)) & 1u)) {
      const unsigned short* r = sm + (size_t)i * 128;
      const unsigned int add =
          (unsigned int)r[2 * t] | ((unsigned int)r[2 * t + 1] << 16);
      removed[t] |= add;
    }
    __syncthreads();
  }
  keepSorted[mi * 64 + t] = ~removed[t];
}

// ---------------------------------------------------------------------------
// Kernel 5: combine keepL & keepR & valid, find the 100th kept score
// (threshold; ties pass, matching ms >= kth), write all rows of the output.
// ---------------------------------------------------------------------------
__global__ void finalize_kernel(const int* __restrict__ order,
                                const unsigned int* __restrict__ keepSorted,
                                const int* __restrict__ valid,
                                const float* __restrict__ score,
                                const float4* __restrict__ lbox,
                                const float4* __restrict__ rbox,
                                float* __restrict__ out) {
  __shared__ unsigned char kept[NPROP];
  __shared__ float kthVal;
  __shared__ int hasKth;
  const int img = blockIdx.x;
  const int t = threadIdx.x;  // 0..255
  const int* ord = order + (size_t)img * NPROP;
  const unsigned int* kL = keepSorted + (img * 2 + 0) * 64;
  const unsigned int* kR = keepSorted + (img * 2 + 1) * 64;

  for (int p = t; p < NPROP; p += 256) {
    const int idx = ord[p];
    const bool kk = (((kL[p >> 5] >> (p & 31)) & 1u) != 0u) &&
                    (((kR[p >> 5] >> (p & 31)) & 1u) != 0u) &&
                    (valid[(size_t)img * NPROP + idx] != 0);
    kept[p] = kk ? 1 : 0;
  }
  __syncthreads();

  if (t == 0) {
    int cnt = 0;
    hasKth = 0;
    kthVal = 0.0f;
    for (int p = 0; p < NPROP; ++p) {
      if (kept[p]) {
        ++cnt;
        if (cnt == DETS) {
          hasKth = 1;
          kthVal = score[(size_t)img * NPROP + ord[p]];
          break;
        }
      }
    }
  }
  __syncthreads();

  for (int p = t; p < NPROP; p += 256) {
    const int idx = ord[p];
    const size_t g = (size_t)img * NPROP + idx;
    const float sc = score[g];
    const bool fk = kept[p] && (!hasKth || sc >= kthVal);
    const float mm = fk ? 1.0f : 0.0f;
    const float4 lb = lbox[g];
    const float4 rb = rbox[g];
    float* o = out + g * 9;
    o[0] = lb.x * mm; o[1] = lb.y * mm; o[2] = lb.z * mm; o[3] = lb.w * mm;
    o[4] = rb.x * mm; o[5] = rb.y * mm; o[6] = rb.z * mm; o[7] = rb.w * mm;
    o[8] = fk ? sc : 0.0f;
  }
}

// ---------------------------------------------------------------------------
extern "C" void kernel_launch(void* const* d_in, const int* in_sizes, int n_in,
                              void* d_out, int out_size, void* d_ws,
                              size_t ws_size, hipStream_t stream) {
  const float* logits = (const float*)d_in[0];        // [8192, 81]
  const float* reg = (const float*)d_in[1];           // [8192, 648]
  const float4* lprop = (const float4*)d_in[2];       // [8192, 4]
  const float4* rprop = (const float4*)d_in[3];       // [8192, 4]
  float* out = (float*)d_out;                         // [4, 2048, 9]

  char* w = (char*)d_ws;
  float4* lbox = (float4*)w;                 w += (size_t)ROWS * sizeof(float4);
  float4* rbox = (float4*)w;                 w += (size_t)ROWS * sizeof(float4);
  float4* sortedL = (float4*)w;              w += (size_t)ROWS * sizeof(float4);
  float4* sortedR = (float4*)w;              w += (size_t)ROWS * sizeof(float4);
  float* score = (float*)w;                  w += (size_t)ROWS * sizeof(float);
  float* skey = (float*)w;                   w += (size_t)ROWS * sizeof(float);
  int* valid = (int*)w;                      w += (size_t)ROWS * sizeof(int);
  int* order = (int*)w;                      w += (size_t)ROWS * sizeof(int);
  unsigned int* keepSorted = (unsigned int*)w;
  w += (size_t)BATCH * 2 * 64 * sizeof(unsigned int);
  // align to 16B
  w = (char*)(((uintptr_t)w + 15u) & ~(uintptr_t)15u);
  unsigned short* supp = (unsigned short*)w;  // 8 * 2048 * 128 u16 = 4 MB

  decode_score_kernel<<<ROWS / 8, 256, 0, stream>>>(
      logits, reg, lprop, rprop, lbox, rbox, score, skey, valid);

  sort_kernel<<<BATCH, 1024, 0, stream>>>(skey, lbox, rbox, order, sortedL,
                                          sortedR);

  supp_kernel<<<dim3(BATCH * 2, 16), 256, 0, stream>>>(sortedL, sortedR, supp);

  nms_scan_kernel<<<BATCH * 2, 64, 0, stream>>>(supp, keepSorted);

  finalize_kernel<<<BATCH, 256, 0, stream>>>(order, keepSorted, valid, score,
                                             lbox, rbox, out);
}